// EpisodicLearningSystem_70068096467217
// MI455X (gfx1250) — compile-verified
//
#include <hip/hip_runtime.h>
#include <hip/hip_bf16.h>
#include <math.h>

// Problem constants (fixed by the reference).
#define BB 256     // batch
#define KK 256     // key dim
#define VV 256     // value dim
#define NN 4000    // memory slots
#define MM 4000    // memory_index length
#define TOPK 8
#define EPS 1e-8f

typedef __attribute__((ext_vector_type(2))) float v2f;
typedef __attribute__((ext_vector_type(8))) float v8f;

// ---------------------------------------------------------------------------
// Kernel 1: e = tanh(Q @ W^T + b), fp32 WMMA 16x16x4. One wave per 16x16 tile.
// A frag (16x4): lane l: m=l%16, K pair = 2*(l/16). B frag (4x16): B[k][n]=W[n][k].
// ---------------------------------------------------------------------------
__global__ __launch_bounds__(32)
void qproj_tanh_kernel(const float* __restrict__ Q, const float* __restrict__ W,
                       const float* __restrict__ bias, float* __restrict__ e) {
    const int lane = threadIdx.x;        // 0..31, wave32
    const int half = lane >> 4;          // 0/1 -> K sub-pair
    const int lm   = lane & 15;
    const int row0 = blockIdx.y * 16;    // batch tile
    const int col0 = blockIdx.x * 16;    // output-feature tile
    v8f c = {};
    for (int k0 = 0; k0 < KK; k0 += 4) {
        const int ka = k0 + 2 * half;
        v2f a, b;
        a.x = Q[(row0 + lm) * KK + ka];
        a.y = Q[(row0 + lm) * KK + ka + 1];
        // e[b][j] = sum_k Q[b][k] * W[j][k]  => B[k][n] = W[col0+n][k]
        b.x = W[(col0 + lm) * KK + ka];
        b.y = W[(col0 + lm) * KK + ka + 1];
        c = __builtin_amdgcn_wmma_f32_16x16x4_f32(false, a, false, b, (short)0, c,
                                                  false, false);
    }
    const float bj = bias[col0 + lm];
#pragma unroll
    for (int v = 0; v < 8; ++v) {
        const int m = v + 8 * half;      // C/D layout: VGPR v, lanes16-31 => M=v+8
        e[(row0 + m) * KK + col0 + lm] = tanhf(c[v] + bj);
    }
}

// ---------------------------------------------------------------------------
// Kernel 2: qn = e / max(||e||, eps), one block per row.
// ---------------------------------------------------------------------------
__global__ __launch_bounds__(256)
void qnorm_kernel(const float* __restrict__ e, float* __restrict__ qn) {
    const int b = blockIdx.x, tid = threadIdx.x;
    __shared__ float red[256];
    const float v = e[b * KK + tid];
    red[tid] = v * v;
    __syncthreads();
    for (int s = 128; s > 0; s >>= 1) {
        if (tid < s) red[tid] += red[tid + s];
        __syncthreads();
    }
    const float inv = 1.0f / fmaxf(sqrtf(red[0]), EPS);
    qn[b * KK + tid] = v * inv;
}

// ---------------------------------------------------------------------------
// Kernel 3: stored_keys[m] = assoc[memory_index[m], :, 0]; kn = normalize rows.
// One block per m; lane tid reads element k=tid (stride V floats).
// ---------------------------------------------------------------------------
__global__ __launch_bounds__(256)
void kn_kernel(const float* __restrict__ assoc, const int* __restrict__ mem_idx,
               float* __restrict__ kn) {
    const int m = blockIdx.x, tid = threadIdx.x;
    __shared__ float red[256];
    const size_t n = (size_t)mem_idx[m];
    const float v = assoc[n * (size_t)(KK * VV) + (size_t)tid * VV];  // [n, tid, 0]
    red[tid] = v * v;
    __syncthreads();
    for (int s = 128; s > 0; s >>= 1) {
        if (tid < s) red[tid] += red[tid + s];
        __syncthreads();
    }
    const float inv = 1.0f / fmaxf(sqrtf(red[0]), EPS);
    kn[m * KK + tid] = v * inv;
}

// ---------------------------------------------------------------------------
// Kernel 4: sims = qn (256xK) @ kn^T (Kx4000), fp32 WMMA 16x16x4.
// One wave per 16x16 tile; both 16xK operand slabs staged in LDS.
// ---------------------------------------------------------------------------
__global__ __launch_bounds__(32)
void sims_kernel(const float* __restrict__ qn, const float* __restrict__ kn,
                 float* __restrict__ sims) {
    __shared__ float As[16 * KK];
    __shared__ float Bs[16 * KK];
    const int lane = threadIdx.x;
    const int row0 = blockIdx.y * 16;    // b tile
    const int col0 = blockIdx.x * 16;    // m tile
    const float4* q4 = (const float4*)(qn + (size_t)row0 * KK);
    const float4* k4 = (const float4*)(kn + (size_t)col0 * KK);
    float4* A4 = (float4*)As;
    float4* B4 = (float4*)Bs;
    for (int i = lane; i < 16 * KK / 4; i += 32) {
        A4[i] = q4[i];
        B4[i] = k4[i];
    }
    __syncthreads();
    const int half = lane >> 4;
    const int lm   = lane & 15;
    v8f c = {};
    for (int k0 = 0; k0 < KK; k0 += 4) {
        const int ka = k0 + 2 * half;
        v2f a, b;
        a.x = As[lm * KK + ka];
        a.y = As[lm * KK + ka + 1];
        b.x = Bs[lm * KK + ka];          // B[k][n] = kn[col0+n][k]
        b.y = Bs[lm * KK + ka + 1];
        c = __builtin_amdgcn_wmma_f32_16x16x4_f32(false, a, false, b, (short)0, c,
                                                  false, false);
    }
#pragma unroll
    for (int v = 0; v < 8; ++v) {
        const int m = v + 8 * half;
        sims[(size_t)(row0 + m) * MM + col0 + lm] = c[v];
    }
}

// ---------------------------------------------------------------------------
// Kernel 5: per-row top-8 (value desc, ties -> lowest index, matching
// jax.lax.top_k), confidences, write top_sims/conf to d_out, actual ids to ws.
// ---------------------------------------------------------------------------
__global__ __launch_bounds__(256)
void topk_kernel(const float* __restrict__ sims, const float* __restrict__ usage,
                 const int* __restrict__ mem_idx, float* __restrict__ out_conf,
                 float* __restrict__ out_topsims, int* __restrict__ out_actual) {
    const int b = blockIdx.x, tid = threadIdx.x;
    __shared__ float vals[MM];          // 16 KB
    __shared__ float rmax[256];
    __shared__ int   rarg[256];
    for (int i = tid; i < MM; i += 256) vals[i] = sims[(size_t)b * MM + i];
    __syncthreads();
    for (int kk = 0; kk < TOPK; ++kk) {
        float best = -INFINITY;
        int   barg = MM;
        for (int i = tid; i < MM; i += 256) {   // ascending i: '>' keeps lowest idx
            const float v = vals[i];
            if (v > best) { best = v; barg = i; }
        }
        rmax[tid] = best;
        rarg[tid] = barg;
        __syncthreads();
        for (int s = 128; s > 0; s >>= 1) {
            if (tid < s) {
                const float ov = rmax[tid + s];
                const int   oa = rarg[tid + s];
                if (ov > rmax[tid] || (ov == rmax[tid] && oa < rarg[tid])) {
                    rmax[tid] = ov;
                    rarg[tid] = oa;
                }
            }
            __syncthreads();
        }
        if (tid == 0) {
            const int   idx = rarg[0];
            const float ts  = rmax[0];
            const int   act = mem_idx[idx];
            out_topsims[b * TOPK + kk] = ts;
            out_conf[b * TOPK + kk]    = ts * (1.0f + log1pf(usage[act]));
            out_actual[b * TOPK + kk]  = act;
            vals[idx] = -INFINITY;
        }
        __syncthreads();
    }
}

// ---------------------------------------------------------------------------
// Kernel 6 (bandwidth-dominant, ~512 MB streamed): retrieved[b,k,v] =
// sum_d e[b,d] * assoc[actual[b,k], d, v]. One block per (b,k); float4 loads.
// 256 threads = 64 v-quads x 4 d-subrows, LDS reduce across subrows.
// ---------------------------------------------------------------------------
__global__ __launch_bounds__(256)
void einsum_kernel(const float* __restrict__ e, const int* __restrict__ actual,
                   const float* __restrict__ assoc, float* __restrict__ out) {
    const int bk  = blockIdx.x;          // 0..2047
    const int b   = bk >> 3;
    const int tid = threadIdx.x;
    const int q   = tid & 63;            // v quad index (v = 4*q..4*q+3)
    const int dr  = tid >> 6;            // d sub-row 0..3
    __shared__ float  ev[KK];
    __shared__ float4 part[4][64];       // 4 KB
    ev[tid] = e[b * KK + tid];
    __syncthreads();
    const size_t base = (size_t)actual[bk] * (size_t)(KK * VV);
    const float4* A = (const float4*)(assoc + base);  // row d = 64 float4
    float4 acc = {0.f, 0.f, 0.f, 0.f};
    for (int d = dr; d < KK; d += 4) {
        const float4 a = A[d * (VV / 4) + q];
        const float  w = ev[d];
        acc.x = fmaf(w, a.x, acc.x);
        acc.y = fmaf(w, a.y, acc.y);
        acc.z = fmaf(w, a.z, acc.z);
        acc.w = fmaf(w, a.w, acc.w);
    }
    part[dr][q] = acc;
    __syncthreads();
    if (dr == 0) {
        float4 s = part[0][q];
        const float4 p1 = part[1][q], p2 = part[2][q], p3 = part[3][q];
        s.x += p1.x + p2.x + p3.x;
        s.y += p1.y + p2.y + p3.y;
        s.z += p1.z + p2.z + p3.z;
        s.w += p1.w + p2.w + p3.w;
        ((float4*)(out + (size_t)bk * VV))[q] = s;
    }
}

// ---------------------------------------------------------------------------
// Launch. Inputs: 0=query_key(B,K) 1=W_k(K,K) 2=b_k(K) 3=assoc(N,K,V)
// 4=usage(N) 5=memory_index(M) int32 6=top_k scalar.
// Output: retrieved(256*8*256) ++ confidences(256*8) ++ top_sims(256*8), f32.
// ---------------------------------------------------------------------------
extern "C" void kernel_launch(void* const* d_in, const int* in_sizes, int n_in,
                              void* d_out, int out_size, void* d_ws, size_t ws_size,
                              hipStream_t stream) {
    const float* Q     = (const float*)d_in[0];
    const float* W     = (const float*)d_in[1];
    const float* bias  = (const float*)d_in[2];
    const float* assoc = (const float*)d_in[3];
    const float* usage = (const float*)d_in[4];
    const int*   midx  = (const int*)d_in[5];
    (void)in_sizes; (void)n_in; (void)out_size; (void)ws_size;

    // Workspace layout (floats): e, qn, kn, sims, actual(int)
    float* ws    = (float*)d_ws;
    float* e     = ws;                                   // 65536
    float* qn    = e + (size_t)BB * KK;                  // 65536
    float* kn    = qn + (size_t)BB * KK;                 // 1,024,000
    float* sims  = kn + (size_t)MM * KK;                 // 1,024,000
    int*   act   = (int*)(sims + (size_t)BB * MM);       // 2048 ints (~8.7 MB total)

    float* out_ret  = (float*)d_out;                     // (256,8,256)
    float* out_conf = out_ret + (size_t)BB * TOPK * VV;  // (256,8)
    float* out_ts   = out_conf + (size_t)BB * TOPK;      // (256,8)

    qproj_tanh_kernel<<<dim3(KK / 16, BB / 16), 32, 0, stream>>>(Q, W, bias, e);
    qnorm_kernel<<<BB, 256, 0, stream>>>(e, qn);
    kn_kernel<<<MM, 256, 0, stream>>>(assoc, midx, kn);
    sims_kernel<<<dim3(MM / 16, BB / 16), 32, 0, stream>>>(qn, kn, sims);
    topk_kernel<<<BB, 256, 0, stream>>>(sims, usage, midx, out_conf, out_ts, act);
    einsum_kernel<<<BB * TOPK, 256, 0, stream>>>(e, act, assoc, out_ret);
}